// Recompose_43052752175485
// MI455X (gfx1250) — compile-verified
//
#include <hip/hip_runtime.h>

typedef __attribute__((ext_vector_type(2))) float v2f;
typedef __attribute__((ext_vector_type(8))) float v8f;

static constexpr int KDIM = 128;  // reference K

// One wave computes out[n0..n0+15] = targets[n0..n0+15, :] . w  using
// V_WMMA_F32_16X16X4_F32 (exact f32 semantics, no precision narrowing).
//
// A (16x4 f32, 2 VGPRs): lanes 0-15 hold rows m=lane with K={k0,k0+1};
//                        lanes 16-31 hold rows m=lane-16 with K={k0+2,k0+3}
//   -> per lane: global_load_b64 of targets[row, k0+2h .. k0+2h+1], h=lane>>4
// B (4x16 f32, 2 VGPRs): VGPR0 lanes0-15 = row K=k0 (cols=lane), lanes16-31 = K=k0+2;
//                        VGPR1 = K=k0+1 / K=k0+3. We broadcast w into ALL columns
//   -> every column of D equals the wanted dot product; no masking required.
// C/D (16x16 f32, 8 VGPRs): VGPR v: lanes0-15 -> M=v, lanes16-31 -> M=v+8.
//   -> lane 0 holds rows n0..n0+7, lane 16 holds rows n0+8..n0+15.
__global__ __launch_bounds__(256) void recompose_wmma_kernel(
    const float* __restrict__ targets,
    const float* __restrict__ temporal_bases,
    const float* __restrict__ scaling,
    const int* __restrict__ batch_indices,
    float* __restrict__ out,
    int n_tiles)
{
  __shared__ float w[KDIM];
  const int tid = threadIdx.x;
  if (tid < KDIM) {
    const int idx = batch_indices[1];          // only column 1 of the einsum survives
    w[tid] = temporal_bases[(size_t)idx * KDIM + tid] * scaling[tid];
  }
  __syncthreads();

  const int lane = tid & 31;
  const int wave = tid >> 5;
  const int h    = lane >> 4;   // which K sub-pair this half-wave supplies
  const int m    = lane & 15;   // row within the 16-row tile
  const int koff = 2 * h;

  const int wavesPerBlock = blockDim.x >> 5;
  for (int tile = blockIdx.x * wavesPerBlock + wave; tile < n_tiles;
       tile += gridDim.x * wavesPerBlock) {
    const float* arow = targets + (size_t)(tile * 16 + m) * KDIM + koff;
    v8f c0 = {};
    v8f c1 = {};
#pragma unroll
    for (int k0 = 0; k0 < KDIM; k0 += 8) {
      // streaming, zero-reuse data: non-temporal loads
      v2f a0 = __builtin_nontemporal_load((const v2f*)(arow + k0));
      v2f a1 = __builtin_nontemporal_load((const v2f*)(arow + k0 + 4));
      v2f b0 = *(const v2f*)(w + k0 + koff);        // LDS broadcast (bank-conflict free)
      v2f b1 = *(const v2f*)(w + k0 + 4 + koff);
      // 8 args: (neg_a, A, neg_b, B, c_mod, C, reuse_a, reuse_b)
      c0 = __builtin_amdgcn_wmma_f32_16x16x4_f32(false, a0, false, b0,
                                                 (short)0, c0, false, false);
      c1 = __builtin_amdgcn_wmma_f32_16x16x4_f32(false, a1, false, b1,
                                                 (short)0, c1, false, false);
    }
    const v8f c = c0 + c1;
    if (m == 0) {                                   // lanes 0 and 16 hold the 16 results
      const int n0 = tile * 16 + 8 * h;
      *(float4*)(out + n0)     = make_float4(c[0], c[1], c[2], c[3]);
      *(float4*)(out + n0 + 4) = make_float4(c[4], c[5], c[6], c[7]);
    }
  }
}

// Scalar tail for rows beyond the last full 16-row tile (not hit for N=500000).
__global__ void recompose_tail_kernel(
    const float* __restrict__ targets,
    const float* __restrict__ temporal_bases,
    const float* __restrict__ scaling,
    const int* __restrict__ batch_indices,
    float* __restrict__ out,
    int n_start, int n_total)
{
  const int n = n_start + blockIdx.x * blockDim.x + threadIdx.x;
  if (n >= n_total) return;
  const int idx = batch_indices[1];
  float acc = 0.f;
  for (int k = 0; k < KDIM; ++k)
    acc += targets[(size_t)n * KDIM + k] *
           (scaling[k] * temporal_bases[(size_t)idx * KDIM + k]);
  out[n] = acc;
}

extern "C" void kernel_launch(void* const* d_in, const int* in_sizes, int n_in,
                              void* d_out, int out_size, void* d_ws, size_t ws_size,
                              hipStream_t stream) {
  const float* targets = (const float*)d_in[0];   // (N, K) f32
  const float* tb      = (const float*)d_in[1];   // (T, K) f32
  const float* sc      = (const float*)d_in[2];   // (K,)  f32
  const int*   bi      = (const int*)d_in[3];     // (B,)  i32
  float* out = (float*)d_out;                     // (N, 1) f32

  const int N = in_sizes[0] / KDIM;
  const int n_tiles = N / 16;

  if (n_tiles > 0) {
    const int waves_per_block = 8;                // 256 threads = 8 wave32
    const int blocks = (n_tiles + waves_per_block - 1) / waves_per_block;
    recompose_wmma_kernel<<<blocks, 256, 0, stream>>>(targets, tb, sc, bi, out,
                                                      n_tiles);
  }
  const int tail_start = n_tiles * 16;
  if (tail_start < N) {
    const int tail = N - tail_start;
    recompose_tail_kernel<<<(tail + 255) / 256, 256, 0, stream>>>(
        targets, tb, sc, bi, out, tail_start, N);
  }
}